// GATClassifier_3324304687695
// MI455X (gfx1250) — compile-verified
//
#include <hip/hip_runtime.h>
#include <hip/hip_bf16.h>

#define N_NODES 50000
#define N_EDGES 800000
#define IN_C 256
#define HID 64
#define HEADS 4
#define HC 256
#define NUM_CLASSES 10
#define LRELU 0.2f
#define LN_EPS 1e-5f

typedef __attribute__((ext_vector_type(2))) float v2f;
typedef __attribute__((ext_vector_type(8))) float v8f;

__device__ __forceinline__ float warp_sum(float v) {
#pragma unroll
  for (int o = 16; o > 0; o >>= 1) v += __shfl_xor(v, o, 32);
  return v;
}

// float atomic-max via signed/unsigned int atomics (works with -inf init)
__device__ __forceinline__ void atomic_max_float(float* addr, float val) {
  if (val >= 0.0f)
    atomicMax((int*)addr, __float_as_int(val));
  else
    atomicMin((unsigned int*)addr, __float_as_uint(val));
}

// ---------------------------------------------------------------------------
// init: zero accumulators / esum, amax = -inf
// ---------------------------------------------------------------------------
__global__ void k_init(float* __restrict__ acc, float* __restrict__ esum,
                       int* __restrict__ amax) {
  size_t i = (size_t)blockIdx.x * blockDim.x + threadIdx.x;
  size_t stride = (size_t)gridDim.x * blockDim.x;
  const size_t total_acc = (size_t)N_NODES * HC;
  for (size_t j = i; j < total_acc; j += stride) acc[j] = 0.0f;
  const size_t total_h = (size_t)N_NODES * HEADS;
  for (size_t j = i; j < total_h; j += stride) {
    esum[j] = 0.0f;
    amax[j] = (int)0xFF800000;  // -inf
  }
}

// ---------------------------------------------------------------------------
// cedge[h] = sum_c W_edge[h*64+c] * att_edge[h*64+c]   (per-head scalar)
// ---------------------------------------------------------------------------
__global__ void k_cedge(const float* __restrict__ W_edge,
                        const float* __restrict__ att_edge,
                        float* __restrict__ cedge) {
  __shared__ float sm[HC];
  int t = threadIdx.x;
  sm[t] = W_edge[t] * att_edge[t];
  __syncthreads();
  if (t < HEADS) {
    float s = 0.0f;
    for (int c = 0; c < HID; ++c) s += sm[t * HID + c];
    cedge[t] = s;
  }
}

// ---------------------------------------------------------------------------
// xh = x @ W  via V_WMMA_F32_16X16X4_F32.
// One wave -> 16 rows x 32 cols (two 16x16 tiles sharing the A fragment,
// two independent accumulator chains for WMMA pipelining).
// A 16x4 (MxK): lane m=lane&15, VGPR0/1 hold K = 2*(lane>=16) + {0,1}
// B 4x16 (KxN): lane n=lane&15, VGPR0/1 hold K = 2*(lane>=16) + {0,1}
// C 16x16: VGPR v, lanes 0-15 -> M=v, lanes 16-31 -> M=v+8; N = lane&15
// N_NODES = 3125*16 exactly -> no store guards needed.
// ---------------------------------------------------------------------------
__global__ __launch_bounds__(256) void k_gemm_xh(const float* __restrict__ x,
                                                 const float* __restrict__ W,
                                                 float* __restrict__ xh) {
  const int lane = threadIdx.x & 31;
  const int wave = blockIdx.x * 8 + (threadIdx.x >> 5);
  const int rowTile = wave >> 3;  // 8 column-pair tiles span the 256 cols
  const int colPair = wave & 7;
  const int row0 = rowTile * 16;
  const int col0 = colPair * 32;
  const int m = lane & 15;
  const int half = lane >> 4;
  const int koff = half * 2;

  const float* __restrict__ xrow = x + (size_t)(row0 + m) * IN_C;
  const float* __restrict__ wc0 = W + (size_t)(col0 + m);
  const float* __restrict__ wc1 = W + (size_t)(col0 + 16 + m);

  v8f acc0 = {}, acc1 = {};
#pragma unroll 4
  for (int k = 0; k < IN_C; k += 4) {
    v2f a, b0, b1;
    a[0] = xrow[k + koff];
    a[1] = xrow[k + koff + 1];
    size_t r0 = (size_t)(k + koff) * HC;
    size_t r1 = (size_t)(k + koff + 1) * HC;
    b0[0] = wc0[r0];
    b0[1] = wc0[r1];
    b1[0] = wc1[r0];
    b1[1] = wc1[r1];
    acc0 = __builtin_amdgcn_wmma_f32_16x16x4_f32(false, a, false, b0, (short)0,
                                                 acc0, false, false);
    acc1 = __builtin_amdgcn_wmma_f32_16x16x4_f32(false, a, false, b1, (short)0,
                                                 acc1, false, false);
  }
#pragma unroll
  for (int v = 0; v < 8; ++v) {
    int r = row0 + v + 8 * half;
    float* __restrict__ p = xh + (size_t)r * HC + m;
    p[col0] = acc0[v];
    p[col0 + 16] = acc1[v];
  }
}

// ---------------------------------------------------------------------------
// a_src[n,h] / a_dst[n,h] : per-head dot of xh row with attention vectors
// one wave per node; lane covers c = k*32+lane, head = k>>1
// ---------------------------------------------------------------------------
__global__ __launch_bounds__(256) void k_attn_node(
    const float* __restrict__ xh, const float* __restrict__ att_src,
    const float* __restrict__ att_dst, float* __restrict__ a_src,
    float* __restrict__ a_dst) {
  const int lane = threadIdx.x & 31;
  const int n = blockIdx.x * 8 + (threadIdx.x >> 5);
  if (n >= N_NODES) return;
  const float* __restrict__ row = xh + (size_t)n * HC;
  float ss[HEADS] = {0, 0, 0, 0}, sd[HEADS] = {0, 0, 0, 0};
#pragma unroll
  for (int k = 0; k < 8; ++k) {
    int c = k * 32 + lane;
    float v = row[c];
    ss[k >> 1] += v * att_src[c];
    sd[k >> 1] += v * att_dst[c];
  }
#pragma unroll
  for (int h = 0; h < HEADS; ++h) {
    float s1 = warp_sum(ss[h]);
    float s2 = warp_sum(sd[h]);
    if (lane == 0) {
      a_src[(size_t)n * HEADS + h] = s1;
      a_dst[(size_t)n * HEADS + h] = s2;
    }
  }
}

// ---------------------------------------------------------------------------
// pass A: alpha = leaky_relu(a_src[s]+a_dst[d]+ew*cedge); atomicMax amax[dst]
// ---------------------------------------------------------------------------
__global__ __launch_bounds__(256) void k_edge_alpha(
    const long long* __restrict__ ei, const float* __restrict__ ew,
    const float* __restrict__ a_src, const float* __restrict__ a_dst,
    const float* __restrict__ cedge, float4* __restrict__ alpha,
    float* __restrict__ amax) {
  int e = blockIdx.x * blockDim.x + threadIdx.x;
  if (e >= N_EDGES) return;
  long long s = ei[e];
  long long d = ei[N_EDGES + e];
  float w = ew[e];
  float al[HEADS];
#pragma unroll
  for (int h = 0; h < HEADS; ++h) {
    float a = a_src[(size_t)s * HEADS + h] + a_dst[(size_t)d * HEADS + h] +
              w * cedge[h];
    a = (a > 0.0f) ? a : LRELU * a;
    al[h] = a;
    atomic_max_float(&amax[(size_t)d * HEADS + h], a);
  }
  alpha[e] = make_float4(al[0], al[1], al[2], al[3]);
}

// ---------------------------------------------------------------------------
// pass B: ex = exp(alpha - amax[dst]) in place; atomicAdd esum[dst]
// ---------------------------------------------------------------------------
__global__ __launch_bounds__(256) void k_edge_exp(
    const long long* __restrict__ ei, float4* __restrict__ alpha,
    const float* __restrict__ amax, float* __restrict__ esum) {
  int e = blockIdx.x * blockDim.x + threadIdx.x;
  if (e >= N_EDGES) return;
  long long d = ei[N_EDGES + e];
  float4 al = alpha[e];
  float ex[HEADS];
  ex[0] = expf(al.x - amax[(size_t)d * HEADS + 0]);
  ex[1] = expf(al.y - amax[(size_t)d * HEADS + 1]);
  ex[2] = expf(al.z - amax[(size_t)d * HEADS + 2]);
  ex[3] = expf(al.w - amax[(size_t)d * HEADS + 3]);
#pragma unroll
  for (int h = 0; h < HEADS; ++h) atomicAdd(&esum[(size_t)d * HEADS + h], ex[h]);
  alpha[e] = make_float4(ex[0], ex[1], ex[2], ex[3]);
}

// ---------------------------------------------------------------------------
// pass C: scatter messages. One wave per edge; xh and acc are L2-resident.
// ---------------------------------------------------------------------------
__global__ __launch_bounds__(256) void k_message(
    const long long* __restrict__ ei, const float* __restrict__ ex4,
    const float* __restrict__ esum, const float* __restrict__ xh,
    float* __restrict__ acc) {
  const int lane = threadIdx.x & 31;
  const int e = blockIdx.x * 8 + (threadIdx.x >> 5);
  if (e >= N_EDGES) return;
  long long s = ei[e];
  long long d = ei[N_EDGES + e];
  float attn_l = 0.0f;
  if (lane < HEADS)
    attn_l = ex4[(size_t)e * HEADS + lane] /
             (esum[(size_t)d * HEADS + lane] + 1e-16f);
  float attn[HEADS];
#pragma unroll
  for (int h = 0; h < HEADS; ++h) attn[h] = __shfl(attn_l, h, 32);
  const float* __restrict__ srow = xh + (size_t)s * HC;
  float* __restrict__ drow = acc + (size_t)d * HC;
#pragma unroll
  for (int k = 0; k < 8; ++k) {
    int c = k * 32 + lane;
    atomicAdd(&drow[c], srow[c] * attn[k >> 1]);
  }
}

// ---------------------------------------------------------------------------
// pass D: +bias, LayerNorm, ReLU, linear head. One wave per node.
// ---------------------------------------------------------------------------
__global__ __launch_bounds__(256) void k_finalize(
    const float* __restrict__ acc, const float* __restrict__ bias,
    const float* __restrict__ gamma, const float* __restrict__ beta,
    const float* __restrict__ fcW, const float* __restrict__ fcb,
    float* __restrict__ out) {
  const int lane = threadIdx.x & 31;
  const int n = blockIdx.x * 8 + (threadIdx.x >> 5);
  if (n >= N_NODES) return;
  const float* __restrict__ row = acc + (size_t)n * HC;
  float y[8];
  float s = 0.0f;
#pragma unroll
  for (int k = 0; k < 8; ++k) {
    int c = k * 32 + lane;
    y[k] = row[c] + bias[c];
    s += y[k];
  }
  float mu = warp_sum(s) * (1.0f / HC);
  float v = 0.0f;
#pragma unroll
  for (int k = 0; k < 8; ++k) {
    float dlt = y[k] - mu;
    v += dlt * dlt;
  }
  float var = warp_sum(v) * (1.0f / HC);
  float rstd = rsqrtf(var + LN_EPS);
#pragma unroll
  for (int k = 0; k < 8; ++k) {
    int c = k * 32 + lane;
    float t = (y[k] - mu) * rstd * gamma[c] + beta[c];
    y[k] = t > 0.0f ? t : 0.0f;
  }
#pragma unroll
  for (int j = 0; j < NUM_CLASSES; ++j) {
    float p = 0.0f;
#pragma unroll
    for (int k = 0; k < 8; ++k) {
      int c = k * 32 + lane;
      p += y[k] * fcW[(size_t)c * NUM_CLASSES + j];
    }
    p = warp_sum(p);
    if (lane == 0) out[(size_t)n * NUM_CLASSES + j] = p + fcb[j];
  }
}

// ---------------------------------------------------------------------------
extern "C" void kernel_launch(void* const* d_in, const int* in_sizes, int n_in,
                              void* d_out, int out_size, void* d_ws,
                              size_t ws_size, hipStream_t stream) {
  (void)in_sizes; (void)n_in; (void)out_size; (void)ws_size;
  const float* x        = (const float*)d_in[0];
  const long long* ei   = (const long long*)d_in[1];
  const float* ew       = (const float*)d_in[2];
  const float* W        = (const float*)d_in[3];
  const float* att_src  = (const float*)d_in[4];
  const float* att_dst  = (const float*)d_in[5];
  const float* att_edge = (const float*)d_in[6];
  const float* W_edge   = (const float*)d_in[7];
  const float* bias     = (const float*)d_in[8];
  const float* gamma    = (const float*)d_in[9];
  const float* beta     = (const float*)d_in[10];
  const float* fcW      = (const float*)d_in[11];
  const float* fcb      = (const float*)d_in[12];
  float* out = (float*)d_out;

  char* ws = (char*)d_ws;
  size_t off = 0;
  auto walloc = [&](size_t bytes) -> void* {
    void* p = ws + off;
    off += (bytes + 255) & ~(size_t)255;
    return p;
  };
  float* xh    = (float*)walloc((size_t)N_NODES * HC * sizeof(float));
  float* accb  = (float*)walloc((size_t)N_NODES * HC * sizeof(float));
  float* a_src = (float*)walloc((size_t)N_NODES * HEADS * sizeof(float));
  float* a_dst = (float*)walloc((size_t)N_NODES * HEADS * sizeof(float));
  float* amax  = (float*)walloc((size_t)N_NODES * HEADS * sizeof(float));
  float* esum  = (float*)walloc((size_t)N_NODES * HEADS * sizeof(float));
  float* alpha = (float*)walloc((size_t)N_EDGES * HEADS * sizeof(float));
  float* cedge = (float*)walloc(HEADS * sizeof(float));

  k_init<<<2048, 256, 0, stream>>>(accb, esum, (int*)amax);
  k_cedge<<<1, 256, 0, stream>>>(W_edge, att_edge, cedge);

  // 3125 row tiles x 8 col-pair tiles = 25000 waves; 8 waves/block
  int gemm_waves = (N_NODES / 16) * (HC / 32);
  k_gemm_xh<<<gemm_waves / 8, 256, 0, stream>>>(x, W, xh);

  k_attn_node<<<(N_NODES + 7) / 8, 256, 0, stream>>>(xh, att_src, att_dst,
                                                     a_src, a_dst);
  k_edge_alpha<<<(N_EDGES + 255) / 256, 256, 0, stream>>>(
      ei, ew, a_src, a_dst, cedge, (float4*)alpha, amax);
  k_edge_exp<<<(N_EDGES + 255) / 256, 256, 0, stream>>>(ei, (float4*)alpha,
                                                        amax, esum);
  k_message<<<(N_EDGES + 7) / 8, 256, 0, stream>>>(ei, alpha, esum, xh, accb);
  k_finalize<<<(N_NODES + 7) / 8, 256, 0, stream>>>(accb, bias, gamma, beta,
                                                    fcW, fcb, out);
}